// AgentEmbedding_33174327394584
// MI455X (gfx1250) — compile-verified
//
#include <hip/hip_runtime.h>

typedef float v2f __attribute__((ext_vector_type(2)));
typedef float v4f __attribute__((ext_vector_type(4)));
typedef float v8f __attribute__((ext_vector_type(8)));

#define BATCH      64
#define M_AGENTS   50
#define N_CITIES   10000
#define EDIM       128
#define BDIM       128
#define SD_STRIDE  260   // 256 + pad (multiple of 4 for float4 stores, spreads LDS banks)
#define SH_STRIDE  132   // 128 + pad

__global__ __launch_bounds__(BDIM)
void agent_embed_fused(const float* __restrict__ cities,   // [B,N,E]
                       const float* __restrict__ graph,    // [B,1,E]
                       const float* __restrict__ astate,   // [B,M,13]
                       const float* __restrict__ Wdp, const float* __restrict__ bdp,
                       const float* __restrict__ Wdc, const float* __restrict__ bdc,
                       const float* __restrict__ Wnc, const float* __restrict__ bnc,
                       const float* __restrict__ Wps, const float* __restrict__ bps,
                       const float* __restrict__ Wg,  const float* __restrict__ bg,
                       const float* __restrict__ Wa,  const float* __restrict__ ba,
                       float* __restrict__ out)             // [B,M,E]
{
    __shared__ float sD[16 * SD_STRIDE];   // gathered depot rows: 16 x 256
    __shared__ float sH[16 * SH_STRIDE];   // hidden tile: 16 x 128
    __shared__ float sG[EDIM];             // g[b] = Wg*graph + bg
    __shared__ float sBase[EDIM];          // base[b] = Wa[:, :E]*g + ba

    const int tid  = threadIdx.x;
    const int lane = tid & 31;
    const int wave = tid >> 5;
    const int b    = blockIdx.y;
    const int m0   = blockIdx.x * 16;

    // ---- Phase A: per-batch graph embed + its (constant) final-linear part ----
    {
        const int o = tid;                           // one output per thread
        float acc = bg[o];
        const float* gb = graph + (size_t)b * EDIM;
        const float* wr = Wg + (size_t)o * EDIM;
        #pragma unroll 8
        for (int j = 0; j < EDIM; ++j) acc += wr[j] * gb[j];
        sG[o] = acc;
    }
    __syncthreads();
    {
        const int o = tid;
        float acc = ba[o];
        const float* wr = Wa + (size_t)o * (2 * EDIM);   // first half of W_a row
        #pragma unroll 8
        for (int j = 0; j < EDIM; ++j) acc += wr[j] * sG[j];
        sBase[o] = acc;
    }

    // ---- Phase B: gather 16 rows x (2 cities x 128 f32) into LDS via float4 ----
    for (int it = 0; it < 8; ++it) {
        const int u    = tid + it * BDIM;   // 0..1023 float4 transfers
        const int r    = u >> 6;            // agent row in tile, 0..15
        const int half = (u >> 5) & 1;      // which of the two city indices
        const int c4   = u & 31;            // float4 chunk within 128 floats
        const int m    = m0 + r;
        int idx = 0;
        if (m < M_AGENTS)
            idx = (int)astate[((size_t)b * M_AGENTS + m) * 13 + half];
        const float* src = cities + ((size_t)b * N_CITIES + idx) * EDIM + c4 * 4;
        *(v4f*)&sD[r * SD_STRIDE + half * EDIM + c4 * 4] = *(const v4f*)src;
    }

    // ---- Phase C: tiny linears (dc/nc/ps) + all biases -> hidden tile seed ----
    for (int r = 0; r < 16; ++r) {
        const int m = m0 + r;
        float as[11];
        if (m < M_AGENTS) {
            const float* ap = astate + ((size_t)b * M_AGENTS + m) * 13 + 2;
            #pragma unroll
            for (int j = 0; j < 11; ++j) as[j] = ap[j];
        } else {
            #pragma unroll
            for (int j = 0; j < 11; ++j) as[j] = 0.0f;
        }
        const int o = tid;
        float s = bdp[o] + bdc[o] + bnc[o] + bps[o];
        #pragma unroll
        for (int j = 0; j < 4; ++j) s += as[j]     * Wdc[o * 4 + j];
        #pragma unroll
        for (int j = 0; j < 4; ++j) s += as[4 + j] * Wnc[o * 4 + j];
        #pragma unroll
        for (int j = 0; j < 3; ++j) s += as[8 + j] * Wps[o * 3 + j];
        sH[r * SH_STRIDE + o] = s;
    }
    __syncthreads();

    // WMMA fragment geometry (f32 16x16x4, wave32):
    //   A: lane L holds A[L&15][k + 2*(L>=16) + {0,1}] in 2 VGPRs (contiguous float2)
    //   C/D: VGPR r holds row (r + 8*(L>=16)), col (L&15)
    const int hi   = lane >> 4;
    const int nrow = lane & 15;

    // ---- Phase D: depot_pos_embed = depot(16x256) x Wdp^T, accumulate into sH ----
    for (int t = 0; t < 2; ++t) {
        const int col = (wave * 2 + t) * 16 + nrow;   // output column 0..127
        v8f c = {0.f, 0.f, 0.f, 0.f, 0.f, 0.f, 0.f, 0.f};
        const float* wcol = Wdp + (size_t)col * (2 * EDIM);
        for (int k = 0; k < 2 * EDIM; k += 4) {
            v2f a  = *(const v2f*)&sD[nrow * SD_STRIDE + k + 2 * hi];
            v2f bb = *(const v2f*)(wcol + k + 2 * hi);
            c = __builtin_amdgcn_wmma_f32_16x16x4_f32(
                    false, a, false, bb, (short)0, c, false, false);
        }
        #pragma unroll
        for (int r = 0; r < 8; ++r) {
            const int rowM = r + 8 * hi;
            sH[rowM * SH_STRIDE + col] += c[r];   // each element owned by one lane
        }
    }
    __syncthreads();

    // ---- Phase E: out = hidden(16x128) x Wa[:,128:]^T + base ----
    for (int t = 0; t < 2; ++t) {
        const int col = (wave * 2 + t) * 16 + nrow;
        v8f c = {0.f, 0.f, 0.f, 0.f, 0.f, 0.f, 0.f, 0.f};
        const float* wcol = Wa + (size_t)col * (2 * EDIM) + EDIM;  // second half
        for (int k = 0; k < EDIM; k += 4) {
            v2f a  = *(const v2f*)&sH[nrow * SH_STRIDE + k + 2 * hi];
            v2f bb = *(const v2f*)(wcol + k + 2 * hi);
            c = __builtin_amdgcn_wmma_f32_16x16x4_f32(
                    false, a, false, bb, (short)0, c, false, false);
        }
        const float basec = sBase[col];
        #pragma unroll
        for (int r = 0; r < 8; ++r) {
            const int rowM = r + 8 * hi;
            const int m = m0 + rowM;
            if (m < M_AGENTS)
                out[((size_t)b * M_AGENTS + m) * EDIM + col] = c[r] + basec;
        }
    }
}

extern "C" void kernel_launch(void* const* d_in, const int* in_sizes, int n_in,
                              void* d_out, int out_size, void* d_ws, size_t ws_size,
                              hipStream_t stream) {
    (void)in_sizes; (void)n_in; (void)d_ws; (void)ws_size; (void)out_size;
    const float* cities = (const float*)d_in[0];
    const float* graph  = (const float*)d_in[1];
    const float* astate = (const float*)d_in[2];
    const float* Wdp = (const float*)d_in[3];
    const float* bdp = (const float*)d_in[4];
    const float* Wdc = (const float*)d_in[5];
    const float* bdc = (const float*)d_in[6];
    const float* Wnc = (const float*)d_in[7];
    const float* bnc = (const float*)d_in[8];
    const float* Wps = (const float*)d_in[9];
    const float* bps = (const float*)d_in[10];
    const float* Wg  = (const float*)d_in[11];
    const float* bg  = (const float*)d_in[12];
    const float* Wa  = (const float*)d_in[13];
    const float* ba  = (const float*)d_in[14];
    float* out = (float*)d_out;

    dim3 grid(4, BATCH);     // 4 agent tiles of 16, 64 batches
    dim3 block(BDIM);
    agent_embed_fused<<<grid, block, 0, stream>>>(
        cities, graph, astate, Wdp, bdp, Wdc, bdc, Wnc, bnc,
        Wps, bps, Wg, bg, Wa, ba, out);
}